// LightGCN_70128226009643
// MI455X (gfx1250) — compile-verified
//
#include <hip/hip_runtime.h>

typedef float v2f __attribute__((ext_vector_type(2)));
typedef float v8f __attribute__((ext_vector_type(8)));

// -------------------------------------------------------------------------
// Kernel 1: build concat embedding into cur, copy into acc, zero next buffer
// -------------------------------------------------------------------------
__global__ void lgcn_init(const float* __restrict__ ue, const float* __restrict__ ie,
                          float* __restrict__ cur, float* __restrict__ acc,
                          float* __restrict__ nxt, int UD, int ND4) {
  int i = blockIdx.x * blockDim.x + threadIdx.x;
  if (i >= ND4) return;
  int idx = i * 4;
  float4 v;
  if (idx < UD) v = ((const float4*)ue)[i];
  else         v = ((const float4*)ie)[(idx - UD) >> 2];
  ((float4*)cur)[i] = v;
  ((float4*)acc)[i] = v;
  ((float4*)nxt)[i] = make_float4(0.f, 0.f, 0.f, 0.f);
}

// -------------------------------------------------------------------------
// Kernel 2: edge-list SpMM. 64 lanes per edge -> coalesced 256B row gather,
// one relaxed agent-scope f32 atomic add per lane (global_atomic_add_f32).
// Table (38.4MB) is L2-resident on MI455X (192MB L2).
// -------------------------------------------------------------------------
__global__ void lgcn_spmm(const float* __restrict__ vals,
                          const int* __restrict__ src, const int* __restrict__ dst,
                          const float* __restrict__ cur, float* __restrict__ nxt,
                          long long tot) {
  long long t = (long long)blockIdx.x * blockDim.x + threadIdx.x;
  if (t >= tot) return;
  int e = (int)(t >> 6);
  int d = (int)(t & 63);
  if (d == 0) {
    // stream-ahead hint on the edge arrays (global_prefetch_b8)
    __builtin_prefetch(src + e + 2048, 0, 1);
    __builtin_prefetch(dst + e + 2048, 0, 1);
    __builtin_prefetch(vals + e + 2048, 0, 1);
  }
  int s = src[e];
  int q = dst[e];
  float x = vals[e] * cur[(long long)s * 64 + d];
  __hip_atomic_fetch_add(&nxt[(long long)q * 64 + d], x,
                         __ATOMIC_RELAXED, __HIP_MEMORY_SCOPE_AGENT);
}

// -------------------------------------------------------------------------
// Kernel 3: acc += next ; optionally zero the buffer that becomes "next"
// for the following layer (fused single pass, float4 vectorized).
// -------------------------------------------------------------------------
__global__ void lgcn_post(const float* __restrict__ nxt, float* __restrict__ acc,
                          float* __restrict__ zbuf, int ND4, int doZero) {
  int i = blockIdx.x * blockDim.x + threadIdx.x;
  if (i >= ND4) return;
  float4 a = ((float4*)acc)[i];
  float4 n = ((const float4*)nxt)[i];
  a.x += n.x; a.y += n.y; a.z += n.z; a.w += n.w;
  ((float4*)acc)[i] = a;
  if (doZero) ((float4*)zbuf)[i] = make_float4(0.f, 0.f, 0.f, 0.f);
}

// -------------------------------------------------------------------------
// Kernel 4: batched scoring with f32 WMMA.
// One wave handles 16 batch rows. A[i,k] = u[i,k]*v[i,k] (f32, exact),
// B = ones => D = A x B + C accumulates row-sums across K in f32.
// 16 chained V_WMMA_F32_16X16X4_F32 reduce K=64.
// gamma = ub + ib + dot(accU, accI) / 16   (mean-of-4-layers folded in).
// -------------------------------------------------------------------------
__global__ void lgcn_score(const float* __restrict__ accsum,
                           const float* __restrict__ ubias,
                           const float* __restrict__ ibias,
                           const int* __restrict__ users,
                           const int* __restrict__ items,
                           float* __restrict__ out, int U, int B) {
  int wid  = blockIdx.x * (blockDim.x >> 5) + (threadIdx.x >> 5);
  int lane = threadIdx.x & 31;
  int b0 = wid * 16;
  if (b0 >= B) return;                 // uniform per-wave: EXEC stays all-1s

  // A-matrix 16x4 f32 layout: lanes 0-15 -> M=lane, K={0,1}; lanes 16-31 -> M=lane-16, K={2,3}
  int row  = lane & 15;
  int koff = (lane >> 4) * 2;

  int u  = users[b0 + row];
  int it = items[b0 + row];
  const float* up = accsum + (long long)u * 64;
  const float* ip = accsum + ((long long)(U + it)) * 64;

  v8f c = {};                          // 16x16 f32 accumulator (8 VGPRs)
  v2f ones; ones.x = 1.0f; ones.y = 1.0f;

#pragma unroll
  for (int ck = 0; ck < 16; ++ck) {
    int o = ck * 4 + koff;             // even offset -> 8B aligned
    v2f au = *(const v2f*)(up + o);
    v2f av = *(const v2f*)(ip + o);
    v2f a  = au * av;
    // (neg_a, A, neg_b, B, c_mod, C, reuse_a, reuse_b)
    c = __builtin_amdgcn_wmma_f32_16x16x4_f32(false, a, false, ones,
                                              (short)0, c, false, false);
  }

  // D layout: lane 0 VGPR m = D[m,0] (rows 0..7); lane 16 VGPR m = D[8+m,0]
  if (lane == 0 || lane == 16) {
    int base = (lane == 0) ? 0 : 8;
#pragma unroll
    for (int m = 0; m < 8; ++m) {
      int r  = base + m;
      int uu = users[b0 + r];
      int ii = items[b0 + r];
      out[b0 + r] = ubias[uu] + ibias[ii] + c[m] * 0.0625f;
    }
  }
}

// -------------------------------------------------------------------------
// Host launcher
// -------------------------------------------------------------------------
extern "C" void kernel_launch(void* const* d_in, const int* in_sizes, int n_in,
                              void* d_out, int out_size, void* d_ws, size_t ws_size,
                              hipStream_t stream) {
  const float* user_emb  = (const float*)d_in[0];
  const float* item_emb  = (const float*)d_in[1];
  const float* user_bias = (const float*)d_in[2];
  const float* item_bias = (const float*)d_in[3];
  const float* vals      = (const float*)d_in[4];
  const int*   src       = (const int*)d_in[5];
  const int*   dst       = (const int*)d_in[6];
  const int*   users     = (const int*)d_in[7];
  const int*   items     = (const int*)d_in[8];

  const int U = in_sizes[2];           // user_bias is [U,1]
  const int I = in_sizes[3];           // item_bias is [I,1]
  const int D = 64;
  const int E = in_sizes[4];
  const int B = in_sizes[7];
  const int N = U + I;

  const size_t ND = (size_t)N * D;
  float* buf0 = (float*)d_ws;          // e_l (ping)
  float* buf1 = buf0 + ND;             // e_{l+1} (pong)
  float* acc  = buf1 + ND;             // running sum of layer embeddings

  const int ND4 = (int)(ND / 4);
  const int thr = 256;

  lgcn_init<<<(ND4 + thr - 1) / thr, thr, 0, stream>>>(
      user_emb, item_emb, buf0, acc, buf1, U * D, ND4);

  float* cur = buf0;
  float* nxt = buf1;
  const long long tot = (long long)E * 64;
  const int spmmBlocks = (int)((tot + thr - 1) / thr);

  for (int l = 0; l < 3; ++l) {
    lgcn_spmm<<<spmmBlocks, thr, 0, stream>>>(vals, src, dst, cur, nxt, tot);
    lgcn_post<<<(ND4 + thr - 1) / thr, thr, 0, stream>>>(
        nxt, acc, cur, ND4, (l < 2) ? 1 : 0);
    float* tmp = cur; cur = nxt; nxt = tmp;
  }

  // scoring: 16 rows per wave, 8 waves per block -> 128 rows/block
  const int waves  = (B + 15) / 16;
  const int blocks = (waves + 7) / 8;
  lgcn_score<<<blocks, thr, 0, stream>>>(
      acc, user_bias, item_bias, users, items, (float*)d_out, U, B);
}